// ChebySemi_70222715289681
// MI455X (gfx1250) — compile-verified
//
#include <hip/hip_runtime.h>
#include <hip/hip_bf16.h>

// Jacobi smoothing step: out = x + (f - conv3x3(x, k_b)) / 6
// B=64 images, 512x512 f32, one 3x3 kernel per image, zero padding.
// HBM-bound (~201 MB @ 23.3 TB/s => ~9us floor). Strategy:
//   - stage x tiles into LDS with gfx1250 async global->LDS b128 loads (ASYNCcnt)
//   - overlap the f-stream: issue all f b128 loads before the async wait/barrier
//   - 9-tap stencil from LDS; the x-update re-reads x from LDS (no 2nd HBM read)

#define MDIM   512
#define BATCH  64
#define TY     16            // output rows per block
#define LROWS  (TY + 2)      // + top/bottom halo

#if defined(__gfx1250__) && __has_builtin(__builtin_amdgcn_global_load_async_to_lds_b128)
#define USE_ASYNC_LDS 1
#else
#define USE_ASYNC_LDS 0
#endif

// Builtin signature (from hipcc diagnostic): param0 is
// 'int __vector(4) __device__ *'  -> int4 vector pointer in AS(1).
typedef int v4i __attribute__((ext_vector_type(4)));
typedef __attribute__((address_space(1))) v4i gv4i;   // global (prints as __device__)
typedef __attribute__((address_space(3))) v4i lv4i;   // LDS    (prints as __shared__)

__global__ __launch_bounds__(256) void jacobi3x3_step_kernel(
    const float* __restrict__ x, const float* __restrict__ f,
    const float* __restrict__ k, float* __restrict__ out)
{
    __shared__ float lds[LROWS][MDIM];   // 18 * 512 * 4B = 36 KB

    const int tileY = blockIdx.x;        // 0..31
    const int b     = blockIdx.y;        // 0..63
    const int tid   = threadIdx.x;       // 0..255

    const size_t img = (size_t)MDIM * MDIM;
    const float* __restrict__ xb = x + (size_t)b * img;
    const float* __restrict__ fb = f + (size_t)b * img;
    float* __restrict__       ob = out + (size_t)b * img;
    const float* __restrict__ kb = k + b * 9;

    const int gybase = tileY * TY - 1;   // global row of LDS row 0

    // ---- stage LROWS x MDIM f32 into LDS, 16B chunks, zero-pad halo ----
    // 128 chunks per row, LROWS*128 = 2304 chunks, 9 iters of 256 threads.
    // Lanes of a wave cover 32 consecutive chunks of one row -> the
    // in-image test below is wave-uniform.
#pragma unroll
    for (int i = 0; i < (LROWS * MDIM / 4) / 256; ++i) {
        const int e  = i * 256 + tid;     // chunk id
        const int r  = e >> 7;            // LDS row
        const int c4 = (e & 127) * 4;     // float column of chunk
        const int gy = gybase + r;        // global row
        if (gy >= 0 && gy < MDIM) {
#if USE_ASYNC_LDS
            __builtin_amdgcn_global_load_async_to_lds_b128(
                (gv4i*)(xb + (size_t)gy * MDIM + c4),
                (lv4i*)(&lds[r][c4]),
                /*offset=*/0, /*cpol=*/0);
#else
            *(float4*)(&lds[r][c4]) =
                *(const float4*)(xb + (size_t)gy * MDIM + c4);
#endif
        } else {
            float4 z = {0.f, 0.f, 0.f, 0.f};
            *(float4*)(&lds[r][c4]) = z;
        }
    }

    // per-image 3x3 weights (wave-uniform scalar loads)
    const float w00 = kb[0], w01 = kb[1], w02 = kb[2];
    const float w10 = kb[3], w11 = kb[4], w12 = kb[5];
    const float w20 = kb[6], w21 = kb[7], w22 = kb[8];

    // ---- prefetch the f stream while the async x->LDS copies are in flight ----
    const int ty = tid >> 7;             // 0..1
    const int c  = (tid & 127) * 4;      // output column (float4 aligned)

    float4 fv[TY / 2];
#pragma unroll
    for (int p = 0; p < TY / 2; ++p) {
        const int gy = tileY * TY + p * 2 + ty;
        fv[p] = *(const float4*)(fb + (size_t)gy * MDIM + c);
    }

#if USE_ASYNC_LDS
#if __has_builtin(__builtin_amdgcn_s_wait_asynccnt)
    __builtin_amdgcn_s_wait_asynccnt(0);
#else
    asm volatile("s_wait_asynccnt 0" ::: "memory");
#endif
#endif
    __syncthreads();

    // ---- compute: 8 passes x (2 rows x 512 cols), 4 outputs/thread/pass ----
#pragma unroll
    for (int p = 0; p < TY / 2; ++p) {
        const int ry = p * 2 + ty;       // 0..15 within tile
        const int gy = tileY * TY + ry;  // global output row
        const int lr = ry + 1;           // LDS row of this output row

        float acc0 = 0.f, acc1 = 0.f, acc2 = 0.f, acc3 = 0.f;

#pragma unroll
        for (int j = 0; j < 3; ++j) {
            const float* __restrict__ row = &lds[lr - 1 + j][0];
            const float wa = (j == 0) ? w00 : (j == 1) ? w10 : w20;
            const float wb = (j == 0) ? w01 : (j == 1) ? w11 : w21;
            const float wc = (j == 0) ? w02 : (j == 1) ? w12 : w22;

            const float lm1 = (c == 0)         ? 0.f : row[c - 1];
            const float l0  = row[c];
            const float l1  = row[c + 1];
            const float l2  = row[c + 2];
            const float l3  = row[c + 3];
            const float l4  = (c == MDIM - 4)  ? 0.f : row[c + 4];

            acc0 += wa * lm1 + wb * l0 + wc * l1;
            acc1 += wa * l0  + wb * l1 + wc * l2;
            acc2 += wa * l1  + wb * l2 + wc * l3;
            acc3 += wa * l2  + wb * l3 + wc * l4;
        }

        const float xc0 = lds[lr][c];
        const float xc1 = lds[lr][c + 1];
        const float xc2 = lds[lr][c + 2];
        const float xc3 = lds[lr][c + 3];

        const float s = 1.0f / 6.0f;
        float4 ov;
        ov.x = xc0 + (fv[p].x - acc0) * s;
        ov.y = xc1 + (fv[p].y - acc1) * s;
        ov.z = xc2 + (fv[p].z - acc2) * s;
        ov.w = xc3 + (fv[p].w - acc3) * s;
        *(float4*)(ob + (size_t)gy * MDIM + c) = ov;
    }
}

extern "C" void kernel_launch(void* const* d_in, const int* in_sizes, int n_in,
                              void* d_out, int out_size, void* d_ws, size_t ws_size,
                              hipStream_t stream) {
    (void)in_sizes; (void)n_in; (void)out_size; (void)d_ws; (void)ws_size;
    const float* x  = (const float*)d_in[0];   // [64,1,512,512] f32
    const float* f  = (const float*)d_in[1];   // [64,1,512,512] f32
    const float* kA = (const float*)d_in[2];   // [64,1,3,3]    f32
    float* out = (float*)d_out;                // [64,1,512,512] f32

    dim3 grid(MDIM / TY, BATCH);               // 32 x 64 blocks
    dim3 block(256);                           // 8 wave32
    jacobi3x3_step_kernel<<<grid, block, 0, stream>>>(x, f, kA, out);
}